// BlockSparseGraph_29171417874546
// MI455X (gfx1250) — compile-verified
//
#include <hip/hip_runtime.h>
#include <hip/hip_bf16.h>

// BlockSparseGraph scatter-max, MI455X (gfx1250).
//
// Phase 1: zero-fill the 8192x8192 fp32 output (256 MB) with b128 stores.
//          Default (RT) temporal hint on purpose: the dirty lines stay in the
//          192 MB device L2, where phase-2 atomics are executed -> most atomic
//          line fetches hit L2 instead of HBM.
// Phase 2: one edge per lane. Coalesced b32 loads of (w, row, col), then a
//          non-returning GLOBAL_ATOMIC_MAX_I32 on the float bit pattern.
//          With a 0.0f-initialized destination, signed-int max on IEEE bits is
//          bit-exact vs float max (positive floats order like positive ints;
//          negative floats are negative ints and lose to 0 == max(0.0, w)).
//
// No WMMA: this workload has no matrix product; it is HBM/L2 bound
// (~0.3 GB moved -> ~15-25 us at 23.3 TB/s).

__global__ __launch_bounds__(256) void bsg_zero_fill_b128(float4* __restrict__ out,
                                                          unsigned n4) {
  unsigned i = blockIdx.x * 256u + threadIdx.x;
  if (i < n4) {
    out[i] = make_float4(0.0f, 0.0f, 0.0f, 0.0f);
  }
}

__global__ __launch_bounds__(256) void bsg_scatter_max(const float* __restrict__ w,
                                                       const int* __restrict__ rows,
                                                       const int* __restrict__ cols,
                                                       const int* __restrict__ n_ptr,
                                                       int* __restrict__ out,
                                                       int e) {
  int i = blockIdx.x * 256 + threadIdx.x;
  if (i >= e) return;

  // Speculative prefetch of the streamed edge arrays (global_prefetch_b8).
  // OOB addresses are silently dropped per the CDNA5 ISA (speculative TH).
  __builtin_prefetch(&w[i + 16384], 0, 0);
  __builtin_prefetch(&rows[i + 16384], 0, 0);
  __builtin_prefetch(&cols[i + 16384], 0, 0);

  const int n = *n_ptr;  // uniform scalar load (device-held python scalar)
  const float wv = w[i];
  const int r = rows[i];
  const int c = cols[i];
  const long long idx = (long long)r * (long long)n + (long long)c;

  // Non-returning native atomic: signed-int max on float bits == float max
  // given the 0.0f-initialized destination (see header comment).
  atomicMax(&out[idx], __float_as_int(wv));
}

extern "C" void kernel_launch(void* const* d_in, const int* in_sizes, int n_in,
                              void* d_out, int out_size, void* d_ws, size_t ws_size,
                              hipStream_t stream) {
  const float* weights = (const float*)d_in[0];
  const int* rows = (const int*)d_in[1];
  const int* cols = (const int*)d_in[2];
  const int* n_ptr = (const int*)d_in[3];
  const int e = in_sizes[0];

  // out_size = n*n = 67,108,864 for N=8192 (divisible by 4).
  const unsigned n4 = (unsigned)(out_size / 4);
  const unsigned zero_blocks = (n4 + 255u) / 256u;
  bsg_zero_fill_b128<<<zero_blocks, 256, 0, stream>>>((float4*)d_out, n4);

  // Same stream -> fill completes before scatter begins.
  const unsigned scat_blocks = ((unsigned)e + 255u) / 256u;
  bsg_scatter_max<<<scat_blocks, 256, 0, stream>>>(weights, rows, cols, n_ptr,
                                                   (int*)d_out, e);
}